// Decoder_76149770158192
// MI455X (gfx1250) — compile-verified
//
#include <hip/hip_runtime.h>
#include <math.h>

// ---------------------------------------------------------------------------
// Problem constants (from the reference)
// ---------------------------------------------------------------------------
constexpr int cB  = 32;    // batch
constexpr int cT  = 800;   // encoder time steps
constexpr int cL  = 60;    // decoder steps
constexpr int cH  = 512;   // decoder hidden
constexpr int cD  = 1024;  // encoder feature dim (2H)
constexpr int cA  = 512;   // attention hidden
constexpr int cFC = 16;    // attention conv filters
constexpr int cK  = 101;   // attention conv kernel
constexpr int cC  = 4000;  // classes

typedef float v2f __attribute__((ext_vector_type(2)));
typedef float v8f __attribute__((ext_vector_type(8)));

// V_WMMA_F32_16X16X4_F32 : D = A(16x4) x B(4x16) + C(16x16), fp32 end to end.
__device__ __forceinline__ v8f wmma4(v2f a, v2f b, v8f c) {
  return __builtin_amdgcn_wmma_f32_16x16x4_f32(false, a, false, b,
                                               (short)0, c, false, false);
}

// ---------------------------------------------------------------------------
// Generic WMMA GEMM:  out[M,N] (+)= X[M,K] * W[N,K]^T  (+ bias) (tanh)
// One wave computes a 16(M) x (16*Q)(N) strip. Q, ACC, ACT are compile-time:
// the inner loop is straight-line and software-pipelined (next k-step's
// fragments are loaded before this step's WMMAs issue, so the wave keeps a
// full iteration of loads in flight instead of s_wait_loadcnt 0 per WMMA).
// Launcher guarantees (N/16) % Q == 0, M % 16 == 0, K % 8 == 0.
// fp32 16x16x4 fragment layout (ISA 7.12.2):
//   A:  lane l -> row m = l&15, holds K = 2*(l>>4) + {0,1}
//   B:  lane l -> col n = l&15, holds K = 2*(l>>4) + {0,1}  (row-major W)
//   C/D: vgpr r, lane l -> row = 16*mTile + r + 8*(l>>4), col = l&15
// ---------------------------------------------------------------------------
template <int Q, bool ACC, bool ACT>
__global__ void gemm_xwt_wmma(const float* __restrict__ X, int ldx,
                              const float* __restrict__ W, int ldw,
                              const float* __restrict__ bias,
                              float* __restrict__ out, int ldo,
                              int M, int N, int Kd) {
  const int lane    = threadIdx.x & 31;
  const int wave    = blockIdx.x * (blockDim.x >> 5) + (threadIdx.x >> 5);
  const int nGroups = (N >> 4) / Q;
  const int mTiles  = M >> 4;
  if (wave >= mTiles * nGroups) return;          // wave-uniform entry guard
  const int mTile  = wave / nGroups;
  const int nGroup = wave - mTile * nGroups;

  const int m16   = lane & 15;
  const int khalf = (lane >> 4) << 1;            // 0 or 2
  const float* xp = X + (long)((mTile << 4) + m16) * ldx + khalf;
  const int rbase = (mTile << 4) + ((lane >> 4) << 3);

  v8f acc[Q];
  int cols[Q];
  const float* wp[Q];
#pragma unroll
  for (int q = 0; q < Q; ++q) {
    cols[q] = ((nGroup * Q + q) << 4) + m16;
    wp[q]   = W + (long)cols[q] * ldw + khalf;
    if (ACC) {
#pragma unroll
      for (int r = 0; r < 8; ++r)
        acc[q][r] = out[(long)(rbase + r) * ldo + cols[q]];
    } else {
#pragma unroll
      for (int r = 0; r < 8; ++r) acc[q][r] = 0.0f;
    }
  }

  // ---- software-pipelined K loop ---------------------------------------
  v2f a0 = *(const v2f*)(xp);
  v2f b0[Q];
#pragma unroll
  for (int q = 0; q < Q; ++q) b0[q] = *(const v2f*)(wp[q]);

#pragma unroll 2
  for (int k = 4; k < Kd; k += 4) {
    const v2f a1 = *(const v2f*)(xp + k);        // prefetch next fragments
    v2f b1[Q];
#pragma unroll
    for (int q = 0; q < Q; ++q) b1[q] = *(const v2f*)(wp[q] + k);
#pragma unroll
    for (int q = 0; q < Q; ++q)
      acc[q] = wmma4(a0, b0[q], acc[q]);         // consume current fragments
    a0 = a1;
#pragma unroll
    for (int q = 0; q < Q; ++q) b0[q] = b1[q];
  }
#pragma unroll
  for (int q = 0; q < Q; ++q)
    acc[q] = wmma4(a0, b0[q], acc[q]);           // drain

  // ---- epilogue ---------------------------------------------------------
#pragma unroll
  for (int q = 0; q < Q; ++q) {
    const float bv = bias ? bias[cols[q]] : 0.0f;
#pragma unroll
    for (int r = 0; r < 8; ++r) {
      float v = acc[q][r] + bv;
      if (ACT) v = tanhf(v);
      out[(long)(rbase + r) * ldo + cols[q]] = v;
    }
  }
}

// ---------------------------------------------------------------------------
// Location conv: F[b,t,f] = conv_b[f] + sum_k alpha[b,t+k-50] * conv_w[f,0,k]
// ---------------------------------------------------------------------------
__global__ void conv_loc(const float* __restrict__ alpha,
                         const float* __restrict__ cw,
                         const float* __restrict__ cb,
                         float* __restrict__ F) {
  const int idx = blockIdx.x * blockDim.x + threadIdx.x;   // b*T + t
  if (idx >= cB * cT) return;
  const int b = idx / cT;
  const int t = idx - b * cT;
  float accf[cFC];
#pragma unroll
  for (int f = 0; f < cFC; ++f) accf[f] = 0.0f;
  const float* al = alpha + (long)b * cT;
  for (int k = 0; k < cK; ++k) {
    const int tt = t + k - (cK / 2);
    if (tt < 0 || tt >= cT) continue;
    const float av = al[tt];
#pragma unroll
    for (int f = 0; f < cFC; ++f) accf[f] += av * cw[f * cK + k];
  }
#pragma unroll
  for (int f = 0; f < cFC; ++f) F[(long)idx * cFC + f] = accf[f] + cb[f];
}

// ---------------------------------------------------------------------------
// Attention energy (fused): one wave per (b,t)
//   e[b,t] = sum_a w_ee[a]*tanh(sE[b,a] + hE[b,t,a] + sum_f W_fe[a,f]*F[b,t,f])
// then masked with -1e30 where t >= lengths[b].
// ---------------------------------------------------------------------------
__global__ void att_energy(const float* __restrict__ hE,
                           const float* __restrict__ sE,
                           const float* __restrict__ F,
                           const float* __restrict__ Wfe,
                           const float* __restrict__ wee,
                           const int*   __restrict__ lengths,
                           float* __restrict__ e) {
  const int gw   = (blockIdx.x * blockDim.x + threadIdx.x) >> 5;  // (b,t)
  const int lane = threadIdx.x & 31;
  if (gw >= cB * cT) return;                                      // uniform
  const int b = gw / cT;
  const int t = gw - b * cT;

  float Fv[cFC];
  const float* Fp = F + (long)gw * cFC;
#pragma unroll
  for (int f = 0; f < cFC; ++f) Fv[f] = Fp[f];

  const float* hp = hE + (long)gw * cA;
  const float* sp = sE + (long)b * cA;
  float acc = 0.0f;
#pragma unroll 4
  for (int i = 0; i < cA / 32; ++i) {
    const int aIdx = i * 32 + lane;                 // coalesced hE reads
    const float* wf = Wfe + (long)aIdx * cFC;
    float conv = 0.0f;
#pragma unroll
    for (int f = 0; f < cFC; ++f) conv += wf[f] * Fv[f];
    const float v = tanhf(sp[aIdx] + hp[aIdx] + conv);
    acc += wee[aIdx] * v;
  }
  for (int off = 16; off > 0; off >>= 1) acc += __shfl_xor(acc, off, 32);
  if (lane == 0) e[gw] = (t < lengths[b]) ? acc : -1e30f;
}

// ---------------------------------------------------------------------------
// Masked softmax over T, one block per batch element
// ---------------------------------------------------------------------------
__global__ void softmax_T(const float* __restrict__ e, float* __restrict__ alpha) {
  const int b   = blockIdx.x;
  const int tid = threadIdx.x;
  __shared__ float red[256];
  const float* ep = e + (long)b * cT;
  float* ap = alpha + (long)b * cT;

  float m = -3.4e38f;
  for (int t = tid; t < cT; t += 256) m = fmaxf(m, ep[t]);
  red[tid] = m; __syncthreads();
  for (int s = 128; s > 0; s >>= 1) {
    if (tid < s) red[tid] = fmaxf(red[tid], red[tid + s]);
    __syncthreads();
  }
  m = red[0]; __syncthreads();

  float sum = 0.0f;
  for (int t = tid; t < cT; t += 256) {
    const float ex = expf(ep[t] - m);
    ap[t] = ex;
    sum += ex;
  }
  red[tid] = sum; __syncthreads();
  for (int s = 128; s > 0; s >>= 1) {
    if (tid < s) red[tid] += red[tid + s];
    __syncthreads();
  }
  const float inv = 1.0f / red[0];
  for (int t = tid; t < cT; t += 256) ap[t] *= inv;
}

// ---------------------------------------------------------------------------
// Context vector: g[b,d] = sum_t alpha[b,t] * hbatch[b,t,d]
// ---------------------------------------------------------------------------
__global__ void ctx_g(const float* __restrict__ alpha,
                      const float* __restrict__ hb,
                      float* __restrict__ g) {
  const int d = blockIdx.x * blockDim.x + threadIdx.x;
  const int b = blockIdx.y;
  if (d >= cD) return;
  const float* ap = alpha + (long)b * cT;
  const float* hp = hb + (long)b * cT * cD + d;
  float acc = 0.0f;
  for (int t = 0; t < cT; ++t) acc += ap[t] * hp[(long)t * cD];
  g[(long)b * cD + d] = acc;
}

// ---------------------------------------------------------------------------
// LSTM cell. rec holds s@W_ss^T + g@W_gs^T + b_gs;  recA holds the
// precomputed targets @ W_ys^T for all (b,l).  sig(x)=tanh(0.5x)*0.5+0.5
// ---------------------------------------------------------------------------
__global__ void lstm_cell(const float* __restrict__ rec,
                          const float* __restrict__ recA,
                          int l,
                          float* __restrict__ s,
                          float* __restrict__ c) {
  const int i = blockIdx.x * blockDim.x + threadIdx.x;
  if (i >= cB * cH) return;
  const int b = i / cH;
  const int h = i - b * cH;
  const float* rp = rec  + (long)b * 4 * cH;
  const float* ra = recA + ((long)b * cL + l) * 4 * cH;
  const float ig = rp[h]           + ra[h];
  const float fg = rp[cH + h]      + ra[cH + h];
  const float cg = rp[2 * cH + h]  + ra[2 * cH + h];
  const float og = rp[3 * cH + h]  + ra[3 * cH + h];
  const float si = tanhf(ig * 0.5f) * 0.5f + 0.5f;
  const float sf = tanhf(fg * 0.5f) * 0.5f + 0.5f;
  const float so = tanhf(og * 0.5f) * 0.5f + 0.5f;
  const float cn = sf * c[i] + si * tanhf(cg);
  c[i] = cn;
  s[i] = so * tanhf(cn);
}

__global__ void fill_zero(float* __restrict__ p, int n) {
  const int i = blockIdx.x * blockDim.x + threadIdx.x;
  if (i < n) p[i] = 0.0f;
}

// ---------------------------------------------------------------------------
// Host side: pick Q so (N/16) % Q == 0; dispatch template instantiation
// ---------------------------------------------------------------------------
template <int Q, bool ACC, bool ACT>
static inline void launch_gemm_q(const float* X, int ldx, const float* W, int ldw,
                                 const float* bias, float* out, int ldo,
                                 int M, int N, int Kd, hipStream_t st) {
  const int waves   = (M / 16) * ((N / 16) / Q);
  const int threads = waves * 32;
  const int block   = 256;
  const int grid    = (threads + block - 1) / block;
  gemm_xwt_wmma<Q, ACC, ACT><<<grid, block, 0, st>>>(X, ldx, W, ldw, bias,
                                                     out, ldo, M, N, Kd);
}

static inline void launch_gemm(const float* X, int ldx, const float* W, int ldw,
                               const float* bias, float* out, int ldo,
                               int M, int N, int Kd, int acc, int act,
                               hipStream_t st) {
  const int tiles = N / 16;
  if (tiles % 4 == 0) {
    if (acc && act)      launch_gemm_q<4, true,  true >(X, ldx, W, ldw, bias, out, ldo, M, N, Kd, st);
    else if (acc)        launch_gemm_q<4, true,  false>(X, ldx, W, ldw, bias, out, ldo, M, N, Kd, st);
    else if (act)        launch_gemm_q<4, false, true >(X, ldx, W, ldw, bias, out, ldo, M, N, Kd, st);
    else                 launch_gemm_q<4, false, false>(X, ldx, W, ldw, bias, out, ldo, M, N, Kd, st);
  } else if (tiles % 2 == 0) {
    if (acc && act)      launch_gemm_q<2, true,  true >(X, ldx, W, ldw, bias, out, ldo, M, N, Kd, st);
    else if (acc)        launch_gemm_q<2, true,  false>(X, ldx, W, ldw, bias, out, ldo, M, N, Kd, st);
    else if (act)        launch_gemm_q<2, false, true >(X, ldx, W, ldw, bias, out, ldo, M, N, Kd, st);
    else                 launch_gemm_q<2, false, false>(X, ldx, W, ldw, bias, out, ldo, M, N, Kd, st);
  } else {
    if (acc && act)      launch_gemm_q<1, true,  true >(X, ldx, W, ldw, bias, out, ldo, M, N, Kd, st);
    else if (acc)        launch_gemm_q<1, true,  false>(X, ldx, W, ldw, bias, out, ldo, M, N, Kd, st);
    else if (act)        launch_gemm_q<1, false, true >(X, ldx, W, ldw, bias, out, ldo, M, N, Kd, st);
    else                 launch_gemm_q<1, false, false>(X, ldx, W, ldw, bias, out, ldo, M, N, Kd, st);
  }
}

extern "C" void kernel_launch(void* const* d_in, const int* in_sizes, int n_in,
                              void* d_out, int out_size, void* d_ws, size_t ws_size,
                              hipStream_t stream) {
  (void)in_sizes; (void)n_in; (void)out_size; (void)ws_size;

  const float* hbatch  = (const float*)d_in[0];
  const int*   lengths = (const int*)  d_in[1];
  const float* targets = (const float*)d_in[2];
  const float* W_sy    = (const float*)d_in[3];
  const float* W_gy    = (const float*)d_in[4];
  const float* b_gy    = (const float*)d_in[5];
  const float* W_yy    = (const float*)d_in[6];
  const float* b_yy    = (const float*)d_in[7];
  const float* W_ys    = (const float*)d_in[8];
  const float* W_ss    = (const float*)d_in[9];
  const float* W_gs    = (const float*)d_in[10];
  const float* b_gs    = (const float*)d_in[11];
  const float* W_se    = (const float*)d_in[12];
  const float* W_he    = (const float*)d_in[13];
  const float* W_fe    = (const float*)d_in[14];
  const float* conv_w  = (const float*)d_in[15];
  const float* conv_b  = (const float*)d_in[16];
  const float* w_ee    = (const float*)d_in[17];
  float* out = (float*)d_out;

  // Workspace layout (floats), ~71 MB total
  float* ws    = (float*)d_ws;
  float* hE    = ws;                               // [B*T, A]    13,107,200
  float* recA  = hE    + (long)cB * cT * cA;       // [B*L, 4H]    3,932,160
  float* F     = recA  + (long)cB * cL * 4 * cH;   // [B*T, FC]      409,600
  float* e     = F     + (long)cB * cT * cFC;      // [B, T]          25,600
  float* alpha = e     + (long)cB * cT;            // [B, T]          25,600
  float* s     = alpha + (long)cB * cT;            // [B, H]          16,384
  float* c     = s     + (long)cB * cH;            // [B, H]          16,384
  float* sE    = c     + (long)cB * cH;            // [B, A]          16,384
  float* g     = sE    + (long)cB * cA;            // [B, D]          32,768
  float* ytmp  = g     + (long)cB * cD;            // [B, H]          16,384
  float* rec   = ytmp  + (long)cB * cH;            // [B, 4H]         65,536

  // ---- init state -------------------------------------------------------
  fill_zero<<<(cB * cH + 255) / 256, 256, 0, stream>>>(s, cB * cH);
  fill_zero<<<(cB * cH + 255) / 256, 256, 0, stream>>>(c, cB * cH);
  fill_zero<<<(cB * cT + 255) / 256, 256, 0, stream>>>(alpha, cB * cT);

  // ---- precompute (fully parallel, off the sequential critical path) ----
  // hE = hbatch @ W_he^T        (25600 x 512 x 1024)
  launch_gemm(hbatch, cD, W_he, cD, nullptr, hE, cA,
              cB * cT, cA, cD, 0, 0, stream);
  // recA = targets @ W_ys^T     (1920 x 2048 x 4000) — state-independent,
  // hoisted out of the 60-step scan (removes the largest GEMM per step).
  launch_gemm(targets, cC, W_ys, cC, nullptr, recA, 4 * cH,
              cB * cL, 4 * cH, cC, 0, 0, stream);

  // ---- decoder scan -----------------------------------------------------
  for (int l = 0; l < cL; ++l) {
    // location conv on previous alpha
    conv_loc<<<(cB * cT + 255) / 256, 256, 0, stream>>>(alpha, conv_w, conv_b, F);

    // sE = s @ W_se^T
    launch_gemm(s, cH, W_se, cH, nullptr, sE, cA, cB, cA, cH, 0, 0, stream);

    // rec(partial) = s @ W_ss^T   (uses previous-step s; independent of attn)
    launch_gemm(s, cH, W_ss, cH, nullptr, rec, 4 * cH, cB, 4 * cH, cH, 0, 0, stream);

    // energies (fused tanh + reduce + mask)
    att_energy<<<(cB * cT * 32 + 255) / 256, 256, 0, stream>>>(
        hE, sE, F, W_fe, w_ee, lengths, e);

    // alpha = softmax(e)
    softmax_T<<<cB, 256, 0, stream>>>(e, alpha);

    // g = alpha @ hbatch (per batch)
    ctx_g<<<dim3(cD / 256, cB), 256, 0, stream>>>(alpha, hbatch, g);

    // ytmp = tanh(g @ W_gy^T + s @ W_sy^T + b_gy)
    launch_gemm(g, cD, W_gy, cD, nullptr, ytmp, cH, cB, cH, cD, 0, 0, stream);
    launch_gemm(s, cH, W_sy, cH, b_gy,    ytmp, cH, cB, cH, cH, 1, 1, stream);

    // y = ytmp @ W_yy^T + b_yy  -> d_out[:, l, :]  (250 tiles -> Q=2)
    launch_gemm(ytmp, cH, W_yy, cH, b_yy, out + (long)l * cC, cL * cC,
                cB, cC, cH, 0, 0, stream);

    // rec += g @ W_gs^T + b_gs
    launch_gemm(g, cD, W_gs, cD, b_gs, rec, 4 * cH, cB, 4 * cH, cD, 1, 0, stream);

    // LSTM cell update (s, c);  gates = rec + recA[:, l, :]
    lstm_cell<<<(cB * cH + 255) / 256, 256, 0, stream>>>(rec, recA, l, s, c);
  }
}